// feq_WeightGenerator_807453852203
// MI455X (gfx1250) — compile-verified
//
#include <hip/hip_runtime.h>
#include <math.h>

#define B_   4
#define IN_  256
#define OUT_ 256
#define EMB_ 256
#define H_   120
#define W_   160
#define FH_  15
#define FW_  20
#define MED_ 512
#define NF_  6
#define FS_  81
#define HW_  (H_*W_)          /* 19200 */
#define CELLS_ (FH_*FW_)      /* 300 */

typedef __attribute__((ext_vector_type(2))) float v2f;
typedef __attribute__((ext_vector_type(4))) float v4f;
typedef __attribute__((ext_vector_type(8))) float v8f;

__device__ __forceinline__ v8f vzero8(){ v8f z = {0.f,0.f,0.f,0.f,0.f,0.f,0.f,0.f}; return z; }

// ---- wave-level WMMA f32 16x16xK inner loop. LDS tiles:
//   A tile: [16 rows m][ld], row-major over k ; B tile: [16 rows n][ld]
// per-lane mapping (ISA 7.12.2): m/n = lane&15, k = 2*(lane>>4) + vgpr
__device__ __forceinline__ v8f wmma_loop(const float* Al, const float* Bl,
                                         int ld, int kc, v8f acc){
  const int lane = threadIdx.x & 31;
  const int r    = lane & 15;
  const int ko   = (lane >> 4) << 1;
  const float* ap = Al + r*ld + ko;
  const float* bp = Bl + r*ld + ko;
#pragma unroll 4
  for (int k = 0; k < kc; k += 4){
    v2f a = *(const v2f*)(ap + k);
    v2f b = *(const v2f*)(bp + k);
    acc = __builtin_amdgcn_wmma_f32_16x16x4_f32(false, a, false, b, (short)0, acc, false, false);
  }
  return acc;
}

__device__ __forceinline__ void wmma_cplx(const float* Ar, const float* Ai,
                                          const float* Br, const float* Bi,
                                          int ld, int kc,
                                          v8f& rr, v8f& ri, v8f& ir, v8f& ii){
  const int lane = threadIdx.x & 31;
  const int r    = lane & 15;
  const int ko   = (lane >> 4) << 1;
#pragma unroll 2
  for (int k = 0; k < kc; k += 4){
    v2f ar = *(const v2f*)(Ar + r*ld + ko + k);
    v2f ai = *(const v2f*)(Ai + r*ld + ko + k);
    v2f br = *(const v2f*)(Br + r*ld + ko + k);
    v2f bi = *(const v2f*)(Bi + r*ld + ko + k);
    rr = __builtin_amdgcn_wmma_f32_16x16x4_f32(false, ar, false, br, (short)0, rr, false, false);
    ri = __builtin_amdgcn_wmma_f32_16x16x4_f32(false, ar, false, bi, (short)0, ri, false, false);
    ir = __builtin_amdgcn_wmma_f32_16x16x4_f32(false, ai, false, br, (short)0, ir, false, false);
    ii = __builtin_amdgcn_wmma_f32_16x16x4_f32(false, ai, false, bi, (short)0, ii, false, false);
  }
}

// ====================== DFT matrix init (ortho) ======================
__global__ __launch_bounds__(256) void k_init_dft(float* Fr, float* Fi, float* Gr, float* Gi,
                                                  float* Cr, float* Ci){
  int t = blockIdx.x*blockDim.x + threadIdx.x;
  const float TP = 6.283185307179586f;
  const float sW = 0.07905694150420949f;   // 1/sqrt(160)
  const float sH = 0.09128709291752769f;   // 1/sqrt(120)
  if (t < FS_*W_){            // rFFT along W: Fr/Fi [81][160]
    int k = t / W_, x = t % W_;
    float th = TP * (float)((k*x) % W_) / (float)W_;
    Fr[t] =  cosf(th)*sW;  Fi[t] = -sinf(th)*sW;
  }
  if (t < H_*H_){             // FFT along H: Gr/Gi [120][120]
    int k = t / H_, h = t % H_;
    float th = TP * (float)((k*h) % H_) / (float)H_;
    Gr[t] =  cosf(th)*sH;  Gi[t] = -sinf(th)*sH;
  }
  if (t < W_*FS_){            // irFFT along W: Cr/Ci [160][81] (hermitian-folded)
    int n = t / FS_, k = t % FS_;
    float th = TP * (float)((k*n) % W_) / (float)W_;
    float w = (k == 0 || k == FS_-1) ? sW : 2.f*sW;
    Cr[t] =  cosf(th)*w;   Ci[t] = -sinf(th)*w;
  }
}

// ====================== routing (tiny, VALU) ======================
__global__ __launch_bounds__(256) void k_routing(const float* __restrict__ x,
      const float* __restrict__ rw1w, const float* __restrict__ rw1b,
      const float* __restrict__ rw2w, const float* __restrict__ rw2b,
      float* __restrict__ r){
  __shared__ float z[IN_];
  __shared__ float hid[2];
  const int b = blockIdx.x, c = threadIdx.x;
  {
    const float* xp = x + ((size_t)b*IN_ + c)*CELLS_;
    float s = 0.f;
    for (int p = 0; p < CELLS_; ++p) s += xp[p];
    z[c] = s * (1.f/(float)CELLS_);
  }
  __syncthreads();
  if (c < 2){
    float a = rw1b[c];
    for (int i = 0; i < IN_; ++i) a += rw1w[c*IN_ + i]*z[i];
    hid[c] = 0.5f*a*(1.f + erff(a*0.7071067811865476f));   // exact GELU
  }
  __syncthreads();
  for (int m = c; m < MED_; m += 256){
    float raw[NF_]; float mx = -1e30f;
    for (int f = 0; f < NF_; ++f){
      int row = f*MED_ + m;
      raw[f] = hid[0]*rw2w[row*2] + hid[1]*rw2w[row*2+1] + rw2b[row];
      mx = fmaxf(mx, raw[f]);
    }
    float den = 0.f;
    for (int f = 0; f < NF_; ++f){ raw[f] = expf(raw[f]-mx); den += raw[f]; }
    float inv = 1.f/den;
    for (int f = 0; f < NF_; ++f) r[((size_t)b*NF_ + f)*MED_ + m] = raw[f]*inv;
  }
}

// ====================== feature_fuse: 1x1 conv -> dw3x3 -> BN -> StarReLU ==========
__global__ __launch_bounds__(320) void k_ffuse(const float* __restrict__ x,
      const float* __restrict__ w1, const float* __restrict__ b1,
      const float* __restrict__ dw, const float* __restrict__ dwb,
      const float* __restrict__ bg, const float* __restrict__ bb,
      const float* __restrict__ bm, const float* __restrict__ bv,
      const float* __restrict__ ffs, const float* __restrict__ ffb,
      float* __restrict__ xf){
  const int bo = blockIdx.x, b = bo >> 7, o = bo & 127;
  __shared__ float t[FH_*FW_];
  const int p = threadIdx.x;
  if (p < CELLS_){
    const float* xp = x + (size_t)b*IN_*CELLS_ + p;
    const float* wr = w1 + (size_t)o*IN_;
    float a = b1[o];
    for (int c = 0; c < IN_; ++c) a += wr[c]*xp[(size_t)c*CELLS_];
    t[p] = a;
  }
  __syncthreads();
  if (p < CELLS_){
    const int i = p / FW_, j = p % FW_;
    float a = dwb[o];
    const float* k9 = dw + o*9;
    for (int di = -1; di <= 1; ++di)
      for (int dj = -1; dj <= 1; ++dj){
        int ii = i+di, jj = j+dj;
        if (ii >= 0 && ii < FH_ && jj >= 0 && jj < FW_)
          a += k9[(di+1)*3 + (dj+1)] * t[ii*FW_ + jj];
      }
    float inv = bg[o]*rsqrtf(bv[o] + 1e-5f);
    a = (a - bm[o])*inv + bb[o];
    float rl = fmaxf(a, 0.f);
    xf[((size_t)b*128 + o)*CELLS_ + p] = ffs[0]*rl*rl + ffb[0];
  }
}

// ====================== pw1: feq[p][med] = SR(W1 x Yb) ======================
#define P1_KC 64
#define P1_LD 68
__global__ __launch_bounds__(256) void k_pw1(const float* __restrict__ y,
      const float* __restrict__ w1, const float* __restrict__ b1,
      const float* __restrict__ a1s, const float* __restrict__ a1b,
      float* __restrict__ feq, int b){
  __shared__ float Al[64*P1_LD];
  __shared__ float Bl[64*P1_LD];
  const int m0 = blockIdx.x*64;           // med
  const int n0 = blockIdx.y*64;           // pixel
  const float* Y = y + (size_t)b*OUT_*HW_;
  const int tid = threadIdx.x, wv = tid >> 5, lane = tid & 31;
  const int lr = lane & 15, lhi = lane >> 4;
  v8f acc[2]; acc[0] = vzero8(); acc[1] = vzero8();
  for (int k0 = 0; k0 < OUT_; k0 += P1_KC){
    if (k0 + P1_KC < OUT_)
      __builtin_prefetch(Y + (size_t)(k0+P1_KC)*HW_ + n0, 0, 1);
    for (int idx = tid; idx < 64*16; idx += 256){         // A: w1 rows (contig k)
      int row = idx >> 4, q = idx & 15;
      v4f v = *(const v4f*)(w1 + (size_t)(m0+row)*OUT_ + k0 + q*4);
      *(v4f*)(Al + row*P1_LD + q*4) = v;
    }
    for (int idx = tid; idx < 64*16; idx += 256){         // B: y rows, transpose into [n][k]
      int kr = idx >> 4, q = idx & 15;
      v4f v = *(const v4f*)(Y + (size_t)(k0+kr)*HW_ + n0 + q*4);
      Bl[(q*4+0)*P1_LD + kr] = v.x;
      Bl[(q*4+1)*P1_LD + kr] = v.y;
      Bl[(q*4+2)*P1_LD + kr] = v.z;
      Bl[(q*4+3)*P1_LD + kr] = v.w;
    }
    __syncthreads();
    for (int j = 0; j < 2; ++j){
      int job = wv*2 + j, mt = job >> 2, nt = job & 3;
      acc[j] = wmma_loop(Al + mt*16*P1_LD, Bl + nt*16*P1_LD, P1_LD, P1_KC, acc[j]);
    }
    __syncthreads();
  }
  const float s = a1s[0], bo = a1b[0];
  for (int j = 0; j < 2; ++j){
    int job = wv*2 + j, mt = job >> 2, nt = job & 3;
    int n = n0 + nt*16 + lr;
    int mb = m0 + mt*16 + (lhi << 3);
    float* op = feq + (size_t)n*MED_ + mb;
    float t[8];
#pragma unroll
    for (int v = 0; v < 8; ++v){
      float val = acc[j][v] + b1[mb + v];
      float rl = fmaxf(val, 0.f);
      t[v] = s*rl*rl + bo;
    }
    v4f o0 = {t[0],t[1],t[2],t[3]}, o1 = {t[4],t[5],t[6],t[7]};
    *(v4f*)op = o0; *(v4f*)(op+4) = o1;
  }
}

// ====================== rDFT along W: Fw[h][kw][c] = F x feq[h][:][c] ==========
#define DW_KC 40
#define DW_LD 44
__global__ __launch_bounds__(256) void k_dftw(const float* __restrict__ feq,
      const float* __restrict__ Fr, const float* __restrict__ Fi,
      float* __restrict__ Owr, float* __restrict__ Owi){
  __shared__ float Ar[96*DW_LD];
  __shared__ float Ai[96*DW_LD];
  __shared__ float Bl[64*DW_LD];
  const int c0 = blockIdx.x*64, h = blockIdx.y;
  const float* S = feq + (size_t)h*W_*MED_;
  const int tid = threadIdx.x, wv = tid >> 5, lane = tid & 31;
  const int lr = lane & 15, lhi = lane >> 4;
  v8f acc[6];
#pragma unroll
  for (int j = 0; j < 6; ++j) acc[j] = vzero8();
  for (int k0 = 0; k0 < W_; k0 += DW_KC){
    for (int idx = tid; idx < 96*DW_KC; idx += 256){
      int row = idx / DW_KC, k = idx - row*DW_KC;
      float vr = 0.f, vi = 0.f;
      if (row < FS_){ vr = Fr[row*W_ + k0 + k]; vi = Fi[row*W_ + k0 + k]; }
      Ar[row*DW_LD + k] = vr; Ai[row*DW_LD + k] = vi;
    }
    for (int idx = tid; idx < DW_KC*16; idx += 256){
      int k = idx >> 4, q = idx & 15;
      v4f v = *(const v4f*)(S + (size_t)(k0+k)*MED_ + c0 + q*4);
      Bl[(q*4+0)*DW_LD + k] = v.x;
      Bl[(q*4+1)*DW_LD + k] = v.y;
      Bl[(q*4+2)*DW_LD + k] = v.z;
      Bl[(q*4+3)*DW_LD + k] = v.w;
    }
    __syncthreads();
    for (int j = 0; j < 6; ++j){
      int job = wv*6 + j;
      int rem = (job < 24) ? job : job - 24;
      int mt = rem >> 2, nt = rem & 3;
      const float* Ap = ((job < 24) ? Ar : Ai) + mt*16*DW_LD;
      acc[j] = wmma_loop(Ap, Bl + nt*16*DW_LD, DW_LD, DW_KC, acc[j]);
    }
    __syncthreads();
  }
  for (int j = 0; j < 6; ++j){
    int job = wv*6 + j;
    int isIm = (job >= 24);
    int rem = isIm ? job - 24 : job;
    int mt = rem >> 2, nt = rem & 3;
    int n = c0 + nt*16 + lr;
    int mb = mt*16 + (lhi << 3);
    float* O = (isIm ? Owi : Owr) + (size_t)h*FS_*MED_;
#pragma unroll
    for (int v = 0; v < 8; ++v){
      int m = mb + v;
      if (m < FS_) O[(size_t)m*MED_ + n] = acc[j][v];
    }
  }
}

// ====================== complex DFT along H (fwd w/ wgt-mul, or inverse) ==========
#define DH_KC 24
#define DH_LD 28
__global__ __launch_bounds__(256) void k_dfth(const float* __restrict__ Ir,
      const float* __restrict__ Ii, float* __restrict__ Or_, float* __restrict__ Oi_,
      const float* __restrict__ Gr, const float* __restrict__ Gi,
      const float* __restrict__ rbuf, const float* __restrict__ cw,
      int b, int inverse, int domul){
  __shared__ float Agr[128*DH_LD];
  __shared__ float Agi[128*DH_LD];
  __shared__ float Br[64*DH_LD];
  __shared__ float Bi[64*DH_LD];
  const int c0 = blockIdx.x*64, kw = blockIdx.y;
  const int tid = threadIdx.x, wv = tid >> 5, lane = tid & 31;
  const int lr = lane & 15, lhi = lane >> 4;
  v8f arr[4], ari[4], air[4], aii[4];
#pragma unroll
  for (int j = 0; j < 4; ++j){ arr[j]=vzero8(); ari[j]=vzero8(); air[j]=vzero8(); aii[j]=vzero8(); }
  for (int k0 = 0; k0 < H_; k0 += DH_KC){
    for (int idx = tid; idx < 128*DH_KC; idx += 256){
      int row = idx / DH_KC, k = idx - row*DH_KC;
      float vr = 0.f, vi = 0.f;
      if (row < H_){ vr = Gr[row*H_ + k0 + k]; vi = Gi[row*H_ + k0 + k]; }
      Agr[row*DH_LD + k] = vr; Agi[row*DH_LD + k] = vi;
    }
    for (int idx = tid; idx < DH_KC*16; idx += 256){
      int k = idx >> 4, q = idx & 15;
      size_t off = ((size_t)(k0+k)*FS_ + kw)*MED_ + c0 + q*4;
      v4f vr = *(const v4f*)(Ir + off);
      v4f vi = *(const v4f*)(Ii + off);
      Br[(q*4+0)*DH_LD + k] = vr.x; Br[(q*4+1)*DH_LD + k] = vr.y;
      Br[(q*4+2)*DH_LD + k] = vr.z; Br[(q*4+3)*DH_LD + k] = vr.w;
      Bi[(q*4+0)*DH_LD + k] = vi.x; Bi[(q*4+1)*DH_LD + k] = vi.y;
      Bi[(q*4+2)*DH_LD + k] = vi.z; Bi[(q*4+3)*DH_LD + k] = vi.w;
    }
    __syncthreads();
    for (int j = 0; j < 4; ++j){
      int job = wv*4 + j, mt = job >> 2, nt = job & 3;
      wmma_cplx(Agr + mt*16*DH_LD, Agi + mt*16*DH_LD,
                Br + nt*16*DH_LD, Bi + nt*16*DH_LD, DH_LD, DH_KC,
                arr[j], ari[j], air[j], aii[j]);
    }
    __syncthreads();
  }
  for (int j = 0; j < 4; ++j){
    int job = wv*4 + j, mt = job >> 2, nt = job & 3;
    int n = c0 + nt*16 + lr;
    int mb = mt*16 + (lhi << 3);
    float rv[NF_];
    if (domul){
#pragma unroll
      for (int f = 0; f < NF_; ++f) rv[f] = rbuf[((size_t)b*NF_ + f)*MED_ + n];
    }
#pragma unroll
    for (int v = 0; v < 8; ++v){
      int m = mb + v;
      if (m >= H_) continue;
      float Orv = inverse ? (arr[j][v] + aii[j][v]) : (arr[j][v] - aii[j][v]);
      float Oiv = inverse ? (ari[j][v] - air[j][v]) : (ari[j][v] + air[j][v]);
      if (domul){
        const float* cwp = cw + ((size_t)m*FS_ + kw)*(NF_*2);
        float wr = 0.f, wi = 0.f;
#pragma unroll
        for (int f = 0; f < NF_; ++f){ wr += rv[f]*cwp[f*2]; wi += rv[f]*cwp[f*2+1]; }
        float tr = Orv*wr - Oiv*wi;
        float ti = Orv*wi + Oiv*wr;
        Orv = tr; Oiv = ti;
      }
      size_t off = ((size_t)m*FS_ + kw)*MED_ + n;
      Or_[off] = Orv; Oi_[off] = Oiv;
    }
  }
}

// ====================== irDFT along W + StarReLU(act2): feq[h][n][c] ==========
#define IW_KC 28
#define IW_LD 36
__global__ __launch_bounds__(256) void k_irfftw(const float* __restrict__ Xr,
      const float* __restrict__ Xi, const float* __restrict__ Cr, const float* __restrict__ Ci,
      const float* __restrict__ a2s, const float* __restrict__ a2b,
      float* __restrict__ feq){
  __shared__ float Ac[160*IW_LD];
  __shared__ float As[160*IW_LD];
  __shared__ float Br[64*IW_LD];
  __shared__ float Bi[64*IW_LD];
  const int c0 = blockIdx.x*64, h = blockIdx.y;
  const int tid = threadIdx.x, wv = tid >> 5, lane = tid & 31;
  const int lr = lane & 15, lhi = lane >> 4;
  v8f aR[5], aI[5];
#pragma unroll
  for (int j = 0; j < 5; ++j){ aR[j] = vzero8(); aI[j] = vzero8(); }
  for (int k0 = 0; k0 < 84; k0 += IW_KC){
    for (int idx = tid; idx < 160*IW_KC; idx += 256){
      int row = idx / IW_KC, k = idx - row*IW_KC;
      int kk = k0 + k;
      float vr = 0.f, vi = 0.f;
      if (kk < FS_){ vr = Cr[row*FS_ + kk]; vi = Ci[row*FS_ + kk]; }
      Ac[row*IW_LD + k] = vr; As[row*IW_LD + k] = vi;
    }
    for (int idx = tid; idx < IW_KC*16; idx += 256){
      int k = idx >> 4, q = idx & 15;
      int kk = k0 + k;
      v4f vr = {0.f,0.f,0.f,0.f}, vi = {0.f,0.f,0.f,0.f};
      if (kk < FS_){
        size_t off = ((size_t)h*FS_ + kk)*MED_ + c0 + q*4;
        vr = *(const v4f*)(Xr + off);
        vi = *(const v4f*)(Xi + off);
      }
      Br[(q*4+0)*IW_LD + k] = vr.x; Br[(q*4+1)*IW_LD + k] = vr.y;
      Br[(q*4+2)*IW_LD + k] = vr.z; Br[(q*4+3)*IW_LD + k] = vr.w;
      Bi[(q*4+0)*IW_LD + k] = vi.x; Bi[(q*4+1)*IW_LD + k] = vi.y;
      Bi[(q*4+2)*IW_LD + k] = vi.z; Bi[(q*4+3)*IW_LD + k] = vi.w;
    }
    __syncthreads();
    for (int j = 0; j < 5; ++j){
      int job = wv*5 + j, mt = job >> 2, nt = job & 3;
      aR[j] = wmma_loop(Ac + mt*16*IW_LD, Br + nt*16*IW_LD, IW_LD, IW_KC, aR[j]);
      aI[j] = wmma_loop(As + mt*16*IW_LD, Bi + nt*16*IW_LD, IW_LD, IW_KC, aI[j]);
    }
    __syncthreads();
  }
  const float s = a2s[0], bo = a2b[0];
  for (int j = 0; j < 5; ++j){
    int job = wv*5 + j, mt = job >> 2, nt = job & 3;
    int n = c0 + nt*16 + lr;
    int mb = mt*16 + (lhi << 3);
#pragma unroll
    for (int v = 0; v < 8; ++v){
      int m = mb + v;                    // output x position (0..159)
      float val = aR[j][v] + aI[j][v];
      float rl = fmaxf(val, 0.f);
      feq[((size_t)h*W_ + m)*MED_ + n] = s*rl*rl + bo;
    }
  }
}

// ====================== dynamic decoder: per-cell Wd-gen + [256x256]x[256x64] ==========
#define DC_KC 32
#define DC_LD 36
__global__ __launch_bounds__(256) void k_dec(const float* __restrict__ y,
      const float* __restrict__ xf, const float* __restrict__ cew,
      const float* __restrict__ ceb, float* __restrict__ yd, int b){
  __shared__ float Al[256*DC_LD];
  __shared__ float Bl[64*DC_LD];
  __shared__ float xv[128];
  const int cell = blockIdx.x, ci = cell / FW_, cj = cell % FW_;
  const int tid = threadIdx.x, wv = tid >> 5, lane = tid & 31;
  const int lr = lane & 15, lhi = lane >> 4;
  if (tid < 128) xv[tid] = xf[((size_t)b*128 + tid)*CELLS_ + cell];
  __syncthreads();
  v8f acc[8];
#pragma unroll
  for (int j = 0; j < 8; ++j) acc[j] = vzero8();
  const float* Yb = y + (size_t)b*OUT_*HW_ + (size_t)(ci*8)*W_ + cj*8;
  for (int o0 = 0; o0 < OUT_; o0 += DC_KC){
    // generate Wd chunk: Al[e][ol] = ce_b + sum_c xv[g*16+c]*ce_w[e*256+o][c]
    for (int idx = tid; idx < 256*DC_KC; idx += 256){
      int e = idx >> 5, ol = idx & 31;
      int row = e*OUT_ + o0 + ol;
      const float* wp = cew + (size_t)row*16;
      const float* xp = xv + ((e >> 5) << 4);
      float v = ceb[row];
#pragma unroll
      for (int c = 0; c < 16; ++c) v += xp[c]*wp[c];
      Al[e*DC_LD + ol] = v;
    }
    // y tile: Bl[pq][ol]
    for (int idx = tid; idx < DC_KC*64; idx += 256){
      int ol = idx >> 6, pq = idx & 63;
      Bl[pq*DC_LD + ol] = Yb[(size_t)(o0+ol)*HW_ + (pq >> 3)*W_ + (pq & 7)];
    }
    __syncthreads();
    for (int j = 0; j < 8; ++j){
      int job = wv*8 + j, mt = job >> 2, nt = job & 3;
      acc[j] = wmma_loop(Al + mt*16*DC_LD, Bl + nt*16*DC_LD, DC_LD, DC_KC, acc[j]);
    }
    __syncthreads();
  }
  for (int j = 0; j < 8; ++j){
    int job = wv*8 + j, mt = job >> 2, nt = job & 3;
    int pq = nt*16 + lr;
    int eb = mt*16 + (lhi << 3);
    size_t pp = (size_t)(ci*8 + (pq >> 3))*W_ + cj*8 + (pq & 7);
#pragma unroll
    for (int v = 0; v < 8; ++v)
      yd[(size_t)(eb + v)*HW_ + pp] = acc[j][v];
  }
}

// ====================== final: out = yd * (1 + pw2(SR(feq))) ======================
#define FN_KC 32
#define FN_LD 36
__global__ __launch_bounds__(256) void k_final(const float* __restrict__ feq,
      const float* __restrict__ w2, const float* __restrict__ b2,
      const float* __restrict__ yd, float* __restrict__ out, int b){
  __shared__ float Al[64*FN_LD];     // A[m=pixel][k=med]
  __shared__ float Bl[256*FN_LD];    // B[n=emb][k=med]
  const int p0 = blockIdx.x*64;
  const int tid = threadIdx.x, wv = tid >> 5, lane = tid & 31;
  const int lr = lane & 15, lhi = lane >> 4;
  v8f acc[8];
#pragma unroll
  for (int j = 0; j < 8; ++j) acc[j] = vzero8();
  for (int k0 = 0; k0 < MED_; k0 += FN_KC){
    for (int idx = tid; idx < 64*8; idx += 256){
      int row = idx >> 3, q = idx & 7;
      v4f v = *(const v4f*)(feq + (size_t)(p0+row)*MED_ + k0 + q*4);
      *(v4f*)(Al + row*FN_LD + q*4) = v;
    }
    for (int idx = tid; idx < 256*8; idx += 256){
      int row = idx >> 3, q = idx & 7;
      v4f v = *(const v4f*)(w2 + (size_t)row*MED_ + k0 + q*4);
      *(v4f*)(Bl + row*FN_LD + q*4) = v;
    }
    __syncthreads();
    for (int j = 0; j < 8; ++j){
      int job = wv*8 + j, mt = job >> 4, nt = job & 15;
      acc[j] = wmma_loop(Al + mt*16*FN_LD, Bl + nt*16*FN_LD, FN_LD, FN_KC, acc[j]);
    }
    __syncthreads();
  }
  for (int j = 0; j < 8; ++j){
    int job = wv*8 + j, mt = job >> 4, nt = job & 15;
    int e = nt*16 + lr;
    int pb = p0 + mt*16 + (lhi << 3);
    const float* ydp = yd + (size_t)e*HW_ + pb;
    float* op = out + ((size_t)b*EMB_ + e)*HW_ + pb;
    float pb2 = b2[e];
    v4f y0 = *(const v4f*)ydp;
    v4f y1 = *(const v4f*)(ydp + 4);
    float t[8];
    t[0]=y0.x; t[1]=y0.y; t[2]=y0.z; t[3]=y0.w;
    t[4]=y1.x; t[5]=y1.y; t[6]=y1.z; t[7]=y1.w;
#pragma unroll
    for (int v = 0; v < 8; ++v) t[v] = t[v] * (1.f + acc[j][v] + pb2);
    v4f o0 = {t[0],t[1],t[2],t[3]}, o1 = {t[4],t[5],t[6],t[7]};
    *(v4f*)op = o0; *(v4f*)(op+4) = o1;
  }
}

// ============================== launch ==============================
extern "C" void kernel_launch(void* const* d_in, const int* in_sizes, int n_in,
                              void* d_out, int out_size, void* d_ws, size_t ws_size,
                              hipStream_t stream){
  (void)in_sizes; (void)n_in; (void)out_size; (void)ws_size;
  const float* x    = (const float*)d_in[0];
  const float* y    = (const float*)d_in[1];
  const float* pw1w = (const float*)d_in[2];
  const float* pw1b = (const float*)d_in[3];
  const float* pw2w = (const float*)d_in[4];
  const float* pw2b = (const float*)d_in[5];
  const float* rw1w = (const float*)d_in[6];
  const float* rw1b = (const float*)d_in[7];
  const float* rw2w = (const float*)d_in[8];
  const float* rw2b = (const float*)d_in[9];
  const float* a1s  = (const float*)d_in[10];
  const float* a1b  = (const float*)d_in[11];
  const float* a2s  = (const float*)d_in[12];
  const float* a2b  = (const float*)d_in[13];
  const float* ffs  = (const float*)d_in[14];
  const float* ffb  = (const float*)d_in[15];
  const float* cw   = (const float*)d_in[16];
  const float* ff1w = (const float*)d_in[17];
  const float* ff1b = (const float*)d_in[18];
  const float* dww  = (const float*)d_in[19];
  const float* dwb  = (const float*)d_in[20];
  const float* bng  = (const float*)d_in[21];
  const float* bnb  = (const float*)d_in[22];
  const float* bnm  = (const float*)d_in[23];
  const float* bnv  = (const float*)d_in[24];
  const float* cew  = (const float*)d_in[25];
  const float* ceb  = (const float*)d_in[26];
  float* out = (float*)d_out;
  float* W = (float*)d_ws;

  float* rbuf = W;                          // 12288
  float* xf   = W + 12288;                  // 153600
  float* Fr   = W + 165888;                 // 12960
  float* Fi   = Fr + 12960;
  float* Gr   = Fi + 12960;                 // 14400
  float* Gi   = Gr + 14400;
  float* Cr   = Gi + 14400;                 // 12960
  float* Ci   = Cr + 12960;
  float* feq  = W + 246528;                 // 9,830,400 (per-batch [HW][MED])
  float* Fwr  = feq + 9830400;              // 4,976,640 each
  float* Fwi  = Fwr + 4976640;
  float* Ffr  = Fwi + 4976640;
  float* Ffi  = Ffr + 4976640;
  float* ydb  = Ffi + 4976640;              // 4,915,200 (per-batch [EMB][HW])

  k_init_dft<<<57, 256, 0, stream>>>(Fr, Fi, Gr, Gi, Cr, Ci);
  k_routing<<<B_, 256, 0, stream>>>(x, rw1w, rw1b, rw2w, rw2b, rbuf);
  k_ffuse<<<B_*128, 320, 0, stream>>>(x, ff1w, ff1b, dww, dwb,
                                      bng, bnb, bnm, bnv, ffs, ffb, xf);
  for (int b = 0; b < B_; ++b){
    k_pw1<<<dim3(MED_/64, HW_/64), 256, 0, stream>>>(y, pw1w, pw1b, a1s, a1b, feq, b);
    k_dftw<<<dim3(MED_/64, H_), 256, 0, stream>>>(feq, Fr, Fi, Fwr, Fwi);
    k_dfth<<<dim3(MED_/64, FS_), 256, 0, stream>>>(Fwr, Fwi, Ffr, Ffi, Gr, Gi,
                                                   rbuf, cw, b, /*inverse=*/0, /*mul=*/1);
    k_dfth<<<dim3(MED_/64, FS_), 256, 0, stream>>>(Ffr, Ffi, Fwr, Fwi, Gr, Gi,
                                                   rbuf, cw, b, /*inverse=*/1, /*mul=*/0);
    k_irfftw<<<dim3(MED_/64, H_), 256, 0, stream>>>(Fwr, Fwi, Cr, Ci, a2s, a2b, feq);
    k_dec<<<CELLS_, 256, 0, stream>>>(y, xf, cew, ceb, ydb, b);
    k_final<<<HW_/64, 256, 0, stream>>>(feq, pw2w, pw2b, ydb, out, b);
  }
}